// VFELayerMinus_9199819948253
// MI455X (gfx1250) — compile-verified
//
#include <hip/hip_runtime.h>
#include <stdint.h>

typedef __attribute__((ext_vector_type(2))) float v2f;
typedef __attribute__((ext_vector_type(8))) float v8f;

// ---------------------------------------------------------------------------
// Segment key: ((b0*64 + b1)*64 + b2)*64 + b3, fits in 24 bits (< 16M).
// ---------------------------------------------------------------------------
__device__ __forceinline__ uint32_t make_key(const int4 b) {
    return ((((uint32_t)b.x * 64u + (uint32_t)b.y) * 64u + (uint32_t)b.z) * 64u
            + (uint32_t)b.w);
}

// ---------------------------------------------------------------------------
// Kernel 1: zero-init only the touched segment-table entries (<= 256 MB of
// stores instead of clearing the whole 2 GiB table). Races are benign: every
// writer writes identical zeros.
// ---------------------------------------------------------------------------
__global__ __launch_bounds__(256) void seg_init_kernel(
    const int4* __restrict__ idx, float* __restrict__ table,
    uint32_t segMask, int nrows)
{
    int r = blockIdx.x * blockDim.x + threadIdx.x;
    if (r >= nrows) return;
    uint32_t seg = make_key(idx[r]) & segMask;
    float4* p = (float4*)(table + (size_t)seg * 32u);
    const float4 z = make_float4(0.f, 0.f, 0.f, 0.f);
#pragma unroll
    for (int i = 0; i < 8; ++i) p[i] = z;
}

// ---------------------------------------------------------------------------
// Kernel 2: fused dual-GEMM + relu*relu + atomic segment max, using the exact
// fp32 WMMA V_WMMA_F32_16X16X4_F32. One wave32 handles a 16-row tile:
//   A (16x16 f32) = [inputs(10) | mean(6)]   (K-steps of 4)
//   linear path uses K-steps 0..2 (rows 0..9 of W1, step 2 half zero)
//   weight path uses K-steps 2..3 (rows 10..15 -> rows 0..5 of W2)
// -> 10 WMMAs per tile (the block-diagonal zero blocks are skipped entirely).
// x = relu(D_lin + b1) * relu(D_wgt + b2) >= 0, so int atomic max is exact.
// ---------------------------------------------------------------------------
__global__ __launch_bounds__(256) void vfe_wmma_kernel(
    const float* __restrict__ inputs, const float* __restrict__ mean,
    const float* __restrict__ lw, const float* __restrict__ lb,
    const float* __restrict__ ww, const float* __restrict__ wb,
    const int4* __restrict__ idx,
    float* __restrict__ out, float* __restrict__ table,
    uint32_t segMask, int ntiles)
{
    const int lane = threadIdx.x & 31;
    const int lo   = lane & 15;      // N-index within 16-wide tile / A-row M
    const int hi   = lane >> 4;      // half-wave select (K pair / M+8)
    const int wavesPerBlock = blockDim.x >> 5;
    const int wavesTotal = gridDim.x * wavesPerBlock;
    const int waveId = blockIdx.x * wavesPerBlock + (int)(threadIdx.x >> 5);

    // ---- Preload B fragments (shared by every tile this wave processes) ----
    // Linear path: W1 (10x32), K-steps 0..2; kb = 4s + 2*hi.
    v2f bL[2][3];
#pragma unroll
    for (int t = 0; t < 2; ++t) {
        const int j = t * 16 + lo;            // W1 column 0..31
#pragma unroll
        for (int s = 0; s < 3; ++s) {
            const int kb = 4 * s + 2 * hi;
            v2f b;
            b.x = (kb     < 10) ? lw[kb * 32 + j]       : 0.f;
            b.y = (kb + 1 < 10) ? lw[(kb + 1) * 32 + j] : 0.f;
            bL[t][s] = b;
        }
    }
    // Weight path: W2 (6x32) occupies combined K rows 10..15; K-steps 2..3.
    v2f bW[2][2];
#pragma unroll
    for (int t = 0; t < 2; ++t) {
        const int j = t * 16 + lo;            // W2 column 0..31
#pragma unroll
        for (int s = 0; s < 2; ++s) {
            const int kb = 4 * (s + 2) + 2 * hi;   // 8..15
            v2f b;
            b.x = (kb     >= 10) ? ww[(kb - 10)     * 32 + j] : 0.f;
            b.y = (kb + 1 >= 10) ? ww[(kb + 1 - 10) * 32 + j] : 0.f;
            bW[t][s] = b;
        }
    }
    // Per-lane biases for the two column groups this lane produces.
    const float bl0 = lb[lo], bl1 = lb[lo + 16];
    const float bw0 = wb[lo], bw1 = wb[lo + 16];

    for (int tile = waveId; tile < ntiles; tile += wavesTotal) {
        const int row0 = tile * 16;
        const int ra = row0 + lo;             // A-matrix row for this lane

        // ---- A fragments: K pairs (4s+2hi, 4s+2hi+1) never straddle the
        //      inputs(k<10)/mean(k>=10) boundary -> single aligned float2.
        v2f afr[4];
#pragma unroll
        for (int s = 0; s < 4; ++s) {
            const int kb = 4 * s + 2 * hi;
            const float* src = (kb < 10)
                ? (inputs + (size_t)ra * 10 + kb)
                : (mean   + (size_t)ra * 6  + (kb - 10));
            afr[s] = *(const v2f*)src;
        }

        v8f acc0 = {0,0,0,0,0,0,0,0};   // linear, cols  0..15
        v8f acc1 = {0,0,0,0,0,0,0,0};   // linear, cols 16..31
        v8f acc2 = {0,0,0,0,0,0,0,0};   // weight, cols  0..15
        v8f acc3 = {0,0,0,0,0,0,0,0};   // weight, cols 16..31
#pragma unroll
        for (int s = 0; s < 3; ++s) {   // K rows 0..11 (rows 10,11 are zero)
            acc0 = __builtin_amdgcn_wmma_f32_16x16x4_f32(false, afr[s], false, bL[0][s], (short)0, acc0, false, false);
            acc1 = __builtin_amdgcn_wmma_f32_16x16x4_f32(false, afr[s], false, bL[1][s], (short)0, acc1, false, false);
        }
#pragma unroll
        for (int s = 0; s < 2; ++s) {   // K rows 8..15 (rows 8,9 are zero)
            acc2 = __builtin_amdgcn_wmma_f32_16x16x4_f32(false, afr[s + 2], false, bW[0][s], (short)0, acc2, false, false);
            acc3 = __builtin_amdgcn_wmma_f32_16x16x4_f32(false, afr[s + 2], false, bW[1][s], (short)0, acc3, false, false);
        }

        // D layout: VGPR v, this lane -> (row0 + 8*hi + v, col lo / lo+16).
#pragma unroll
        for (int v = 0; v < 8; ++v) {
            const int row = row0 + 8 * hi + v;
            const uint32_t seg = make_key(idx[row]) & segMask;

            float l0 = acc0[v] + bl0; l0 = l0 > 0.f ? l0 : 0.f;
            float l1 = acc1[v] + bl1; l1 = l1 > 0.f ? l1 : 0.f;
            float w0 = acc2[v] + bw0; w0 = w0 > 0.f ? w0 : 0.f;
            float w1 = acc3[v] + bw1; w1 = w1 > 0.f ? w1 : 0.f;
            const float x0 = l0 * w0;
            const float x1 = l1 * w1;

            float* orow = out + (size_t)row * 64;
            orow[lo]      = x0;
            orow[lo + 16] = x1;

            int* tb = (int*)(table + (size_t)seg * 32u);
            atomicMax(tb + lo,      __float_as_int(x0));   // x >= 0: int max OK
            atomicMax(tb + lo + 16, __float_as_int(x1));
        }
    }
}

// ---------------------------------------------------------------------------
// Kernel 2b: scalar tail for rows not covered by full 16-row tiles.
// (N = 2,000,000 is a multiple of 16, so this normally launches nothing.)
// ---------------------------------------------------------------------------
__global__ __launch_bounds__(256) void vfe_tail_kernel(
    const float* __restrict__ inputs, const float* __restrict__ mean,
    const float* __restrict__ lw, const float* __restrict__ lb,
    const float* __restrict__ ww, const float* __restrict__ wb,
    const int4* __restrict__ idx,
    float* __restrict__ out, float* __restrict__ table,
    uint32_t segMask, int rowStart, int nrows)
{
    int r = rowStart + blockIdx.x * blockDim.x + threadIdx.x;
    if (r >= nrows) return;
    float xi[10], mi[6];
#pragma unroll
    for (int k = 0; k < 10; ++k) xi[k] = inputs[(size_t)r * 10 + k];
#pragma unroll
    for (int k = 0; k < 6; ++k)  mi[k] = mean[(size_t)r * 6 + k];
    const uint32_t seg = make_key(idx[r]) & segMask;
    int* tb = (int*)(table + (size_t)seg * 32u);
    for (int j = 0; j < 32; ++j) {
        float lin = lb[j], wgt = wb[j];
        for (int k = 0; k < 10; ++k) lin += xi[k] * lw[k * 32 + j];
        for (int k = 0; k < 6;  ++k) wgt += mi[k] * ww[k * 32 + j];
        lin = lin > 0.f ? lin : 0.f;
        wgt = wgt > 0.f ? wgt : 0.f;
        const float x = lin * wgt;
        out[(size_t)r * 64 + j] = x;
        atomicMax(tb + j, __float_as_int(x));
    }
}

// ---------------------------------------------------------------------------
// Kernel 3: gather the per-key max back into output cols 32..63 (float4 wide).
// ---------------------------------------------------------------------------
__global__ __launch_bounds__(256) void seg_gather_kernel(
    const int4* __restrict__ idx, const float* __restrict__ table,
    float* __restrict__ out, uint32_t segMask, int nrows)
{
    long tid = (long)blockIdx.x * blockDim.x + threadIdx.x;
    int r = (int)(tid >> 3);
    int q = (int)(tid & 7);
    if (r >= nrows) return;
    const uint32_t seg = make_key(idx[r]) & segMask;
    const float4* src = (const float4*)(table + (size_t)seg * 32u) + q;
    float4* dst = (float4*)(out + (size_t)r * 64 + 32) + q;
    *dst = *src;
}

// ---------------------------------------------------------------------------
// Host launcher
// ---------------------------------------------------------------------------
extern "C" void kernel_launch(void* const* d_in, const int* in_sizes, int n_in,
                              void* d_out, int out_size, void* d_ws, size_t ws_size,
                              hipStream_t stream)
{
    const float* inputs = (const float*)d_in[0];   // (N,10)
    const float* mean   = (const float*)d_in[1];   // (N,6)
    const float* lw     = (const float*)d_in[2];   // (10,32)
    const float* lb     = (const float*)d_in[3];   // (32,)
    const float* ww     = (const float*)d_in[4];   // (6,32)
    const float* wb     = (const float*)d_in[5];   // (32,)
    const int4*  idx    = (const int4*)d_in[6];    // (N,4) int32
    float* out = (float*)d_out;                    // (N,64)
    float* table = (float*)d_ws;                   // segment-max table

    const int N = in_sizes[0] / 10;

    // Segment table capacity: power of two, ideally 2^24 (exact keying, 2 GiB).
    const size_t maxSegs = ws_size / (32 * sizeof(float));
    uint32_t cap = 1;
    while (((size_t)cap * 2) <= maxSegs && cap < (1u << 24)) cap <<= 1;
    const uint32_t segMask = cap - 1;

    // 1) zero-init touched table entries
    seg_init_kernel<<<(N + 255) / 256, 256, 0, stream>>>(idx, table, segMask, N);

    // 2) fused WMMA compute + scatter-max (persistent grid-stride over tiles)
    const int ntiles = N / 16;
    if (ntiles > 0) {
        const int wavesPerBlock = 256 / 32;
        int blocks = (ntiles + wavesPerBlock - 1) / wavesPerBlock;
        if (blocks > 4096) blocks = 4096;
        vfe_wmma_kernel<<<blocks, 256, 0, stream>>>(
            inputs, mean, lw, lb, ww, wb, idx, out, table, segMask, ntiles);
    }
    const int tailStart = ntiles * 16;
    if (tailStart < N) {
        vfe_tail_kernel<<<(N - tailStart + 255) / 256, 256, 0, stream>>>(
            inputs, mean, lw, lb, ww, wb, idx, out, table, segMask, tailStart, N);
    }

    // 3) gather seg-max into cols 32..63
    const long total = (long)N * 8;
    seg_gather_kernel<<<(int)((total + 255) / 256), 256, 0, stream>>>(
        idx, table, out, segMask, N);
}